// VectorQuantizer_46239617908958
// MI455X (gfx1250) — compile-verified
//
#include <hip/hip_runtime.h>

typedef float v2f __attribute__((ext_vector_type(2)));
typedef float v8f __attribute__((ext_vector_type(8)));

namespace {
constexpr int Bn = 32, Dd = 256, Hh = 32, Ww = 32;
constexpr int Kk = 1024;
constexpr int HW = Hh * Ww;                 // 1024
constexpr int Nn = Bn * HW;                 // 32768
constexpr int MT = 32;                      // M-rows per block (2 x 16-row WMMA tiles)
constexpr int NBLK = Nn / MT;               // 1024 main blocks
constexpr int ZPAD = 258;                   // LDS row stride (floats): 258%64=2 -> 32 lanes, 32 banks
constexpr size_t ZQ_ELEMS = (size_t)Bn * Dd * HW;   // 8388608
constexpr size_t IDX_OFF  = ZQ_ELEMS;               // indices start (int32 view)
constexpr size_t LOSS_OFF = ZQ_ELEMS + (size_t)Nn;  // scalar loss
constexpr float LOSS_SCALE = 2.0f / ((float)Nn * (float)Dd);  // vq + commitment
}

// ---- codebook squared norms: one block (64 threads) per code row ----
__global__ __launch_bounds__(64) void vq_cbnorm(const float* __restrict__ cb,
                                                float* __restrict__ norms) {
  const int k = blockIdx.x;
  const int t = threadIdx.x;
  const float4* row = (const float4*)(cb + (size_t)k * Dd);
  float4 v = row[t];                       // 64 threads * 4 floats = 256
  float s = v.x * v.x + v.y * v.y + v.z * v.z + v.w * v.w;
  __shared__ float red[64];
  red[t] = s;
  __syncthreads();
  for (int off = 32; off > 0; off >>= 1) {
    if (t < off) red[t] += red[t + off];
    __syncthreads();
  }
  if (t == 0) norms[k] = red[0];
}

// ---- deterministic fixed-order loss reduction over per-block partials ----
__global__ __launch_bounds__(256) void vq_loss_reduce(const float* __restrict__ partials,
                                                      float* __restrict__ lossOut) {
  __shared__ float red[256];
  const int t = threadIdx.x;
  float s = 0.f;
  for (int i = t; i < NBLK; i += 256) s += partials[i];
  red[t] = s;
  __syncthreads();
  for (int off = 128; off > 0; off >>= 1) {
    if (t < off) red[t] += red[t + off];
    __syncthreads();
  }
  if (t == 0) *lossOut = red[0] * LOSS_SCALE;
}

// ---- fused distance GEMM (f32 WMMA, 32xK tile) + argmin + loss + zq writeback ----
__global__ __launch_bounds__(256) void vq_main(const float* __restrict__ z,
                                               const float* __restrict__ cb,
                                               const float* __restrict__ cbn,
                                               float* __restrict__ zq,
                                               int* __restrict__ idxOut,
                                               float* __restrict__ lossPart) {
  __shared__ float zt[MT * ZPAD];                      // 32 z-rows, padded
  __shared__ unsigned long long redK[8 * MT * 16];     // [wave][row][lane16]
  __shared__ int sIdx[MT];
  __shared__ float sLoss[MT];

  const int t = threadIdx.x;
  const int n0 = blockIdx.x * MT;

  // Stage transposed z tile: zt[row][d] = z[b, d, h, w], n = n0+row.
  {
    const int row = t & 31;
    const int n = n0 + row;
    const int b = n >> 10;
    const int hw = n & 1023;
    const float* zb = z + (size_t)b * Dd * HW + hw;
    for (int d = t >> 5; d < Dd; d += 8)
      zt[row * ZPAD + d] = zb[(size_t)d * HW];         // 128B-coalesced across rows
  }
  __syncthreads();

  const int wave = t >> 5;
  const int lane = t & 31;
  const int lane16 = lane & 15;
  const int dk = (lane >= 16) ? 2 : 0;     // A/B K-slot offset per ISA frag layout
  const int mBase = (lane >= 16) ? 8 : 0;  // C/D row base per ISA frag layout

  unsigned long long bestK0[8], bestK1[8];
#pragma unroll
  for (int i = 0; i < 8; ++i) { bestK0[i] = ~0ull; bestK1[i] = ~0ull; }

  const float* zrow0 = &zt[lane16 * ZPAD];         // row-tile 0: M = lane16
  const float* zrow1 = &zt[(16 + lane16) * ZPAD];  // row-tile 1: M = 16 + lane16
  const int kWave = wave * 128;                    // this wave's 128-code stripe

  for (int tile = 0; tile < 8; ++tile) {
    const int kcol = kWave + tile * 16 + lane16;   // this lane's code (B col == D col)
    const float* cbrow = cb + (size_t)kcol * Dd;
    v8f acc0 = {0.f, 0.f, 0.f, 0.f, 0.f, 0.f, 0.f, 0.f};
    v8f acc1 = {0.f, 0.f, 0.f, 0.f, 0.f, 0.f, 0.f, 0.f};
#pragma unroll 8
    for (int d0 = 0; d0 < Dd; d0 += 4) {
      v2f bb = *(const v2f*)(cbrow + d0 + dk);     // one B frag, reused twice
      v2f a0 = *(const v2f*)(zrow0 + d0 + dk);     // LDS, 8B aligned, padded banks
      v2f a1 = *(const v2f*)(zrow1 + d0 + dk);
      acc0 = __builtin_amdgcn_wmma_f32_16x16x4_f32(
          false, a0, false, bb, (short)0, acc0, false, false);
      acc1 = __builtin_amdgcn_wmma_f32_16x16x4_f32(
          false, a1, false, bb, (short)0, acc1, false, false);
    }
    const float cn = cbn[kcol];
    const unsigned int kbits = (unsigned int)kcol;
#pragma unroll
    for (int i = 0; i < 8; ++i) {
      // score = ||e||^2 - 2 z.e  (||z||^2 constant per row; added later for loss)
      float s0 = fmaf(-2.0f, acc0[i], cn);
      float s1 = fmaf(-2.0f, acc1[i], cn);
      unsigned int u0 = __float_as_uint(s0);
      unsigned int u1 = __float_as_uint(s1);
      u0 = (u0 & 0x80000000u) ? ~u0 : (u0 | 0x80000000u);  // monotonic float->u32
      u1 = (u1 & 0x80000000u) ? ~u1 : (u1 | 0x80000000u);
      unsigned long long k0 = ((unsigned long long)u0 << 32) | kbits;
      unsigned long long k1 = ((unsigned long long)u1 << 32) | kbits;
      if (k0 < bestK0[i]) bestK0[i] = k0;          // min key => min score, lowest idx
      if (k1 < bestK1[i]) bestK1[i] = k1;
    }
  }
#pragma unroll
  for (int i = 0; i < 8; ++i) {
    redK[(wave * MT + (mBase + i)) * 16 + lane16] = bestK0[i];
    redK[(wave * MT + (16 + mBase + i)) * 16 + lane16] = bestK1[i];
  }
  __syncthreads();

  // Per-row final reduction over 8 waves x 16 lanes; loss partial.
  if (t < MT) {
    unsigned long long best = ~0ull;
    for (int w = 0; w < 8; ++w)
#pragma unroll
      for (int l = 0; l < 16; ++l) {
        unsigned long long k = redK[(w * MT + t) * 16 + l];
        if (k < best) best = k;
      }
    const int kmin = (int)(best & 0xFFFFFFFFull);
    const unsigned int u = (unsigned int)(best >> 32);
    const float minScore = (u & 0x80000000u) ? __uint_as_float(u & 0x7FFFFFFFu)
                                             : __uint_as_float(~u);
    float zn = 0.f;
    const float2* zr = (const float2*)&zt[t * ZPAD];
    for (int j = 0; j < Dd / 2; ++j) { float2 v = zr[j]; zn += v.x * v.x + v.y * v.y; }
    idxOut[n0 + t] = kmin;
    sIdx[t] = kmin;
    sLoss[t] = zn + minScore;                          // ||z - e||^2 for this row
  }
  __syncthreads();
  if (t == 0) {
    float s = 0.f;
#pragma unroll
    for (int i = 0; i < MT; ++i) s += sLoss[i];
    lossPart[blockIdx.x] = s;                          // deterministic partial
  }

  // zq_out (straight-through forward == quantized codes), layout (B, D, H, W).
  {
    const int row = t & 31;
    const int n = n0 + row;
    const int b = n >> 10;
    const int hw = n & 1023;
    const float* crow = cb + (size_t)sIdx[row] * Dd;   // L2-hot gather
    float* ob = zq + (size_t)b * Dd * HW + hw;
    for (int d = t >> 5; d < Dd; d += 8)
      ob[(size_t)d * HW] = crow[d];                    // 128B-coalesced stores
  }
}

extern "C" void kernel_launch(void* const* d_in, const int* in_sizes, int n_in,
                              void* d_out, int out_size, void* d_ws, size_t ws_size,
                              hipStream_t stream) {
  (void)in_sizes; (void)n_in; (void)out_size; (void)ws_size;
  const float* z  = (const float*)d_in[0];
  const float* cb = (const float*)d_in[1];
  float* out = (float*)d_out;
  float* cbn      = (float*)d_ws;            // Kk floats
  float* lossPart = cbn + Kk;                // NBLK floats
  float* lossOut  = out + LOSS_OFF;
  int*   idxOut   = (int*)(out + IDX_OFF);

  vq_cbnorm<<<Kk, 64, 0, stream>>>(cb, cbn);
  vq_main<<<NBLK, 256, 0, stream>>>(z, cb, cbn, out, idxOut, lossPart);
  vq_loss_reduce<<<1, 256, 0, stream>>>(lossPart, lossOut);
}